// KLDiracVMF_55825984913726
// MI455X (gfx1250) — compile-verified
//
#include <hip/hip_runtime.h>
#include <math.h>

typedef float v2f __attribute__((ext_vector_type(2)));
typedef float v8f __attribute__((ext_vector_type(8)));

#define K_TERMS 700
#define ZDIM 512

// ---------------------------------------------------------------------------
// Kernel 1: lgamma table  lg[k] = lgamma(k+1) + lgamma(k+v+1), v = 255
// ---------------------------------------------------------------------------
__global__ void lg_table_kernel(float* __restrict__ lg) {
  int k = blockIdx.x * blockDim.x + threadIdx.x;
  if (k < K_TERMS) {
    lg[k] = lgammaf((float)(k + 1)) + lgammaf((float)(k + 256));
  }
}

// D = A x ones + C  : every column of D holds the 4-wide row sum of A.
__device__ inline v8f wmma_rowsum(v2f a, v8f acc) {
#if defined(__HIP_DEVICE_COMPILE__)
  const v2f ones = {1.0f, 1.0f};
  return __builtin_amdgcn_wmma_f32_16x16x4_f32(
      /*neg_a=*/false, a, /*neg_b=*/false, ones,
      /*c_mod=*/(short)0, acc, /*reuse_a=*/false, /*reuse_b=*/false);
#else
  return acc;
#endif
}

// ---------------------------------------------------------------------------
// Kernel 2: per-row dot via f32 WMMA + windowed log-space Bessel series
// Block = 256 threads (8 waves). Each wave owns 32 rows (two 16x16 tiles).
// ---------------------------------------------------------------------------
__global__ __launch_bounds__(256) void vmf_kernel(
    const float* __restrict__ mu, const float* __restrict__ kappa,
    const float* __restrict__ wc, const float* __restrict__ lg_g,
    float* __restrict__ out, int n) {
  __shared__ float s_lg[K_TERMS];
  __shared__ float s_acc[8 * 512];  // 8 waves x 2 tiles x 8 vgprs x 32 lanes

  for (int i = threadIdx.x; i < K_TERMS; i += blockDim.x) s_lg[i] = lg_g[i];

  const int lane = threadIdx.x & 31;
  const int wv = threadIdx.x >> 5;
  const int rowBase = blockIdx.x * 256 + wv * 32;
  const int row = rowBase + lane;
  const bool full = (rowBase + 32 <= n);

  float dot = 0.0f;

  if (full) {
    // A-fragment (16x4 f32): lanes 0-15 hold K=0,1 ; lanes 16-31 hold K=2,3
    // for row M = lane&15.  We fold memory-k into WMMA-K with a permutation
    // that is identical for the (A*B elementwise) product, and B = ones makes
    // the K placement irrelevant anyway (row sum).
    const int half = lane >> 4;   // 0/1 -> which 4-float sub-chunk this lane streams
    const int mrow = lane & 15;
    const float* pm1 = mu + (size_t)(rowBase + mrow) * ZDIM + half * 4;
    const float* pw1 = wc + (size_t)(rowBase + mrow) * ZDIM + half * 4;
    const float* pm2 = mu + (size_t)(rowBase + 16 + mrow) * ZDIM + half * 4;
    const float* pw2 = wc + (size_t)(rowBase + 16 + mrow) * ZDIM + half * 4;

    v8f acc1 = {0.f, 0.f, 0.f, 0.f, 0.f, 0.f, 0.f, 0.f};
    v8f acc2 = {0.f, 0.f, 0.f, 0.f, 0.f, 0.f, 0.f, 0.f};

#pragma unroll 2
    for (int kc = 0; kc < ZDIM; kc += 8) {
      float4 a1 = *(const float4*)(pm1 + kc);
      float4 b1 = *(const float4*)(pw1 + kc);
      float4 a2 = *(const float4*)(pm2 + kc);
      float4 b2 = *(const float4*)(pw2 + kc);
      v2f p;
      p.x = a1.x * b1.x; p.y = a1.y * b1.y;
      acc1 = wmma_rowsum(p, acc1);
      p.x = a1.z * b1.z; p.y = a1.w * b1.w;
      acc1 = wmma_rowsum(p, acc1);
      p.x = a2.x * b2.x; p.y = a2.y * b2.y;
      acc2 = wmma_rowsum(p, acc2);
      p.x = a2.z * b2.z; p.y = a2.w * b2.w;
      acc2 = wmma_rowsum(p, acc2);
    }

    float* sa = s_acc + wv * 512;
#pragma unroll
    for (int v = 0; v < 8; ++v) {
      sa[v * 32 + lane] = acc1[v];
      sa[256 + v * 32 + lane] = acc2[v];
    }
  }

  __syncthreads();  // s_acc visible across lanes; s_lg ready

  if (full) {
    // C/D layout: VGPR v, lanes 0-15 -> (M=v, N=lane); lanes 16-31 -> (M=v+8).
    // All columns equal (B = ones), pick one per row.
    const float* sa = s_acc + wv * 512;
    int t = lane >> 4;      // which 16-row tile
    int m = lane & 15;      // row within tile
    dot = (m < 8) ? sa[t * 256 + m * 32 + m]
                  : sa[t * 256 + (m - 8) * 32 + (m + 8)];
  } else if (row < n) {
    const float* pm = mu + (size_t)row * ZDIM;
    const float* pw = wc + (size_t)row * ZDIM;
    float d = 0.0f;
    for (int k = 0; k < ZDIM; ++k) d = fmaf(pm[k], pw[k], d);
    dot = d;
  }

  if (row < n) {
    const float x = kappa[row];
    const float ct = dot * (1.0f / 64.0f);  // cos_theta = dot / RADIUS
    const float L = logf(0.5f * x);         // log(x/2)

    // analytic peak of 2kL - lg[k]:  (k+1)(k+256) = (x/2)^2
    float kr = 0.5f * (sqrtf(65025.0f + x * x) - 257.0f);
    int kpeak = (int)kr + 1;
    kpeak = min(max(kpeak, 0), K_TERMS - 1);
    int klo = max(kpeak - 160, 0);
    int khi = min(kpeak + 192, K_TERMS);

    const float mlog = fmaf(2.0f * (float)kpeak, L, -s_lg[kpeak]);
    float s = 0.0f;
    for (int k = klo; k < khi; ++k) {
      float tl = fmaf(2.0f * (float)k, L, -s_lg[k]);
      s += __expf(tl - mlog);
    }
    const float lse = mlog + logf(s);

    const float log_ive = fmaf(255.0f, L, lse) - x;          // v*log(x/2)+LSE-x
    const float live_k = logf(1e-6f + expf(log_ive));        // log(1e-6+ive)
    const float l3 = live_k + x;                             // log_iv_kappa
    const float l2 = -255.0f * logf(1e-6f + x);
    const float l1 = -x * ct;
    // + d/2*log(2*pi) + d*log(r), added in reference order
    const float losses =
        (((l1 + l2) + l3) + 470.49652900079237f) + 2129.3481386801518f;

    out[row] = losses;
    out[n + row] = l1;
    out[2 * n + row] = l2;
    out[3 * n + row] = l3;
  }
}

// ---------------------------------------------------------------------------
extern "C" void kernel_launch(void* const* d_in, const int* in_sizes, int n_in,
                              void* d_out, int out_size, void* d_ws,
                              size_t ws_size, hipStream_t stream) {
  const float* mu = (const float*)d_in[0];
  const float* kappa = (const float*)d_in[1];
  const float* wc = (const float*)d_in[2];
  float* out = (float*)d_out;
  float* lg = (float*)d_ws;  // 700 floats of scratch
  const int n = in_sizes[1];  // B (kappa element count)

  hipLaunchKernelGGL(lg_table_kernel, dim3((K_TERMS + 255) / 256), dim3(256),
                     0, stream, lg);
  hipLaunchKernelGGL(vmf_kernel, dim3((n + 255) / 256), dim3(256), 0, stream,
                     mu, kappa, wc, lg, out, n);
}